// SinkhornLoss_5480378270081
// MI455X (gfx1250) — compile-verified
//
#include <hip/hip_runtime.h>
#include <math.h>

typedef __attribute__((ext_vector_type(2))) float v2f;
typedef __attribute__((ext_vector_type(8))) float v8f;

#define DDIM   65536
#define GRID   128
#define WAVES  8
#define BLOCK  (WAVES * 32)
#define KPW    (DDIM / (GRID * WAVES))   /* 64 k-values per wave */
#define NPART  GRID

#define EPSV    0.01f
#define INV_EPS 100.0f
#define NITER   100
#define THRESH  0.1f

// ---------------------------------------------------------------------------
// Kernel 1: partial Gram matrix G = X * X^T  (16 x 16, K = 65536)
// Each wave: 64 k-values via 8 iterations of (one b128 load -> two f32 WMMAs).
// Lane L loads float4 X[L%16][k0 + 4*(L/16)]; the identical register pair is
// fed as both A (16x4) and B (4x16) since their lane/vgpr->k maps coincide.
// ---------------------------------------------------------------------------
__global__ void __launch_bounds__(BLOCK)
gram_partial_kernel(const float* __restrict__ X, float* __restrict__ partial) {
  const int lane = threadIdx.x & 31;
  const int wv   = threadIdx.x >> 5;
  const int r    = lane & 15;       // row of X this lane serves (m and n)
  const int hi   = lane >> 4;       // lane half -> high k bit(s)
  const int waveId = blockIdx.x * WAVES + wv;

  const float* p = X + (size_t)r * DDIM + (size_t)waveId * KPW + hi * 4;

  v8f acc = {0.f, 0.f, 0.f, 0.f, 0.f, 0.f, 0.f, 0.f};
#pragma unroll
  for (int it = 0; it < KPW / 8; ++it) {
    float4 f = *(const float4*)(p + it * 8);
    v2f a = {f.x, f.y};
    v2f b = {f.z, f.w};
    // D = A*B + C, A=B=X rows -> Gram accumulation (k-order permutation is
    // irrelevant for a sum over k as long as A and B share registers).
    acc = __builtin_amdgcn_wmma_f32_16x16x4_f32(false, a, false, a, (short)0,
                                                acc, false, false);
    acc = __builtin_amdgcn_wmma_f32_16x16x4_f32(false, b, false, b, (short)0,
                                                acc, false, false);
  }

  // Reduce the 8 wave tiles inside the block via LDS.
  // C/D layout: lane L, vgpr t -> element (M = t + 8*(L>>4), N = L%16).
  __shared__ float red[WAVES * 256];
#pragma unroll
  for (int t = 0; t < 8; ++t) {
    const int M = t + 8 * hi;
    red[wv * 256 + M * 16 + r] = acc[t];
  }
  __syncthreads();

  float s = 0.f;
#pragma unroll
  for (int w = 0; w < WAVES; ++w) s += red[w * 256 + threadIdx.x];
  partial[blockIdx.x * 256 + threadIdx.x] = s;
}

// ---------------------------------------------------------------------------
// Kernel 2: reduce partials -> G -> C, run 100 Sinkhorn iterations (with the
// reference's "freeze after convergence" semantics), emit 10 * sum(pi * C).
// Single 256-thread block; thread tid owns element (i = tid/16, j = tid%16).
// ---------------------------------------------------------------------------
__global__ void __launch_bounds__(256)
sinkhorn_kernel(const float* __restrict__ partial, float* __restrict__ out) {
  const int tid = threadIdx.x;
  const int i = tid >> 4;
  const int j = tid & 15;

  __shared__ float G[256];
  __shared__ float E[256];
  __shared__ float u[16], v[16], un[16], vn[16];
  __shared__ int doneFlag;

  // Reduce NPART partial Gram tiles (coalesced over tid).
  float g = 0.f;
  for (int b = 0; b < NPART; ++b) g += partial[b * 256 + tid];
  G[tid] = g;
  if (tid < 16) { u[tid] = 0.f; v[tid] = 0.f; }
  if (tid == 0) doneFlag = 0;
  __syncthreads();

  // C_ij = G_ii + G_jj - 2 G_ij
  const float Cij = G[i * 16 + i] + G[j * 16 + j] - 2.0f * G[i * 16 + j];
  const float log_mu = logf(1.0f / 16.0f);   // == log_nu

  for (int iter = 0; iter < NITER; ++iter) {
    // u update: row log-sum-exp of M(u, v)
    E[tid] = expf((-Cij + u[i] + v[j]) * INV_EPS);
    __syncthreads();
    if (tid < 16) {
      float s = 0.f;
#pragma unroll
      for (int jj = 0; jj < 16; ++jj) s += E[tid * 16 + jj];
      un[tid] = EPSV * (log_mu - logf(s + 1e-6f)) + u[tid];
    }
    __syncthreads();

    // v update: column log-sum-exp of M(u_new, v)
    E[tid] = expf((-Cij + un[i] + v[j]) * INV_EPS);
    __syncthreads();
    if (tid < 16) {
      float s = 0.f;
#pragma unroll
      for (int ii = 0; ii < 16; ++ii) s += E[ii * 16 + tid];
      vn[tid] = EPSV * (log_mu - logf(s + 1e-6f)) + v[tid];
    }
    __syncthreads();

    // err + freeze logic (matches jnp.where(done, ...) then done |= err<thresh)
    if (tid == 0) {
      float err = 0.f;
#pragma unroll
      for (int k = 0; k < 16; ++k) err += fabsf(un[k] - u[k]);
      if (!doneFlag) {
#pragma unroll
        for (int k = 0; k < 16; ++k) { u[k] = un[k]; v[k] = vn[k]; }
      }
      if (err < THRESH) doneFlag = 1;
    }
    __syncthreads();
  }

  // loss = LOSS_WEIGHT * sum(pi * C), pi = exp(M(u, v))
  E[tid] = expf((-Cij + u[i] + v[j]) * INV_EPS) * Cij;
  __syncthreads();
  if (tid == 0) {
    float s = 0.f;
    for (int t = 0; t < 256; ++t) s += E[t];
    out[0] = 10.0f * s;
  }
}

extern "C" void kernel_launch(void* const* d_in, const int* in_sizes, int n_in,
                              void* d_out, int out_size, void* d_ws, size_t ws_size,
                              hipStream_t stream) {
  const float* preds_S = (const float*)d_in[0];   // [16, 65536] f32; preds_T unused
  float* partial = (float*)d_ws;                  // NPART * 256 floats = 128 KB
  float* out = (float*)d_out;

  gram_partial_kernel<<<GRID, BLOCK, 0, stream>>>(preds_S, partial);
  sinkhorn_kernel<<<1, 256, 0, stream>>>(partial, out);
}